// ProposalTarget_79594333929575
// MI455X (gfx1250) — compile-verified
//
#include <hip/hip_runtime.h>

#define FG_THRESH 0.5f
#define BG_HI     0.5f
#define BG_LO     0.1f
#define BATCH     256
#define FG_MAX    64

typedef __attribute__((ext_vector_type(2))) float v2f;
typedef __attribute__((ext_vector_type(8))) float v8f;

// ---------------------------------------------------------------------------
// Stage A: gt_cls[k] = argmax_c gt_labels[k,c] via WMMA (exact: labels are
// one-hot, so argmax == dot with [0..C-1]). The class-index weight is folded
// into the A fragment (documented 32-bit 16x4 layout: lanes 0-15 = M,
// VGPR v holds K = v + 2*(lane>=16)); B is all-ones, making the result
// invariant to B's lane->(K,N) mapping: D[m,n] = sum_k A[m,k].
// ---------------------------------------------------------------------------
__global__ void gtcls_wmma_kernel(const float* __restrict__ lab, int K, int C,
                                  int* __restrict__ gt_cls) {
  int tid  = threadIdx.x;
  int lane = tid & 31;
  int wid  = tid >> 5;
  int tile = blockIdx.x * (blockDim.x >> 5) + wid;
  int rowBase = tile * 16;
  int row  = rowBase + (lane & 15);
  int kOff = (lane < 16) ? 0 : 2;   // documented A layout: K = v + 2*(lane>=16)

  v8f acc = {};
  v2f b;
  b.x = 1.0f;                       // B == ones: layout-independent
  b.y = 1.0f;
  int chunks = (C + 3) >> 2;
  for (int ch = 0; ch < chunks; ++ch) {
    int c0 = ch * 4 + kOff;
    v2f a;
    a.x = (row < K && (c0 + 0) < C) ? lab[row * C + c0 + 0] * (float)(c0 + 0)
                                    : 0.0f;
    a.y = (row < K && (c0 + 1) < C) ? lab[row * C + c0 + 1] * (float)(c0 + 1)
                                    : 0.0f;
    acc = __builtin_amdgcn_wmma_f32_16x16x4_f32(false, a, false, b,
                                                (short)0, acc, false, false);
  }
  // D layout (documented): VGPR v -> M=v (lanes 0-15), M=v+8 (lanes 16-31).
  if ((lane & 15) == 0) {
    int half = lane >> 4;
#pragma unroll
    for (int j = 0; j < 8; ++j) {
      int r = rowBase + half * 8 + j;
      if (r < K) gt_cls[r] = (int)(acc[j] + 0.5f);
    }
  }
}

// ---------------------------------------------------------------------------
// Stage B: per-roi max IoU + argmax over K gt boxes; gt boxes staged in LDS
// via CDNA5 async global->LDS loads (ASYNCcnt path).
// ---------------------------------------------------------------------------
__global__ void iou_kernel(const float* __restrict__ proposals, int Np,
                           const float* __restrict__ gt, int K,
                           int* __restrict__ assign, int* __restrict__ code) {
#pragma clang fp contract(off)
  __shared__ float4 sgt[512];
  __shared__ float  sAreaB[512];
  int tid = threadIdx.x;

  // Async copy gt boxes (16B each) into LDS.
  for (int k = tid; k < K; k += blockDim.x) {
    unsigned lds = (unsigned)(unsigned long long)(size_t)&sgt[k];
    unsigned long long ga = (unsigned long long)(size_t)(gt + k * 4);
    asm volatile("global_load_async_to_lds_b128 %0, %1, off"
                 :: "v"(lds), "v"(ga) : "memory");
  }
  asm volatile("s_wait_asynccnt 0" ::: "memory");
  __syncthreads();

  for (int k = tid; k < K; k += blockDim.x) {
    float4 b = sgt[k];
    sAreaB[k] = (b.z - b.x + 1.0f) * (b.w - b.y + 1.0f);
  }
  __syncthreads();

  int i = blockIdx.x * blockDim.x + tid;
  int n_total = Np + K;
  if (i >= n_total) return;

  float ax1, ay1, ax2, ay2;
  if (i < Np) {
    const float* r = proposals + (size_t)i * 4;
    ax1 = r[0]; ay1 = r[1]; ax2 = r[2]; ay2 = r[3];
  } else {
    const float* r = gt + (size_t)(i - Np) * 4;
    ax1 = r[0]; ay1 = r[1]; ax2 = r[2]; ay2 = r[3];
  }
  float areaA = (ax2 - ax1 + 1.0f) * (ay2 - ay1 + 1.0f);

  float best = -1.0f;
  int   bi   = 0;
  for (int k = 0; k < K; ++k) {
    float4 b = sgt[k];
    float iw = fmaxf(fminf(ax2, b.z) - fmaxf(ax1, b.x) + 1.0f, 0.0f);
    float ih = fmaxf(fminf(ay2, b.w) - fmaxf(ay1, b.y) + 1.0f, 0.0f);
    float inter = iw * ih;
    float iou = inter / (areaA + sAreaB[k] - inter);
    if (iou > best) { best = iou; bi = k; }   // strict > == first max (jnp)
  }
  assign[i] = bi;
  code[i]   = (best >= FG_THRESH) ? 1
            : ((best < BG_HI && best >= BG_LO) ? 2 : 0);
}

// ---------------------------------------------------------------------------
// Stage C: deterministic ordered selection (single block, streaming scan).
// Reproduces fg_sorted / bg_sorted (ascending-index stable order) including
// the overflow path into non-bg indices.
// ---------------------------------------------------------------------------
__global__ void select_kernel(const int* __restrict__ code, int n_total,
                              int* __restrict__ keep, int* __restrict__ meta) {
  __shared__ int fgList[FG_MAX];
  __shared__ int bgList[BATCH];
  __shared__ int otList[BATCH];
  __shared__ int waveOff[3][32];
  __shared__ int runTot[3];
  int tid = threadIdx.x;
  int lane = tid & 31;
  int wid  = tid >> 5;
  int nWaves = blockDim.x >> 5;
  if (tid < 3) runTot[tid] = 0;
  __syncthreads();

  for (int base = 0; base < n_total; base += blockDim.x) {
    int i = base + tid;
    int c = (i < n_total) ? code[i] : -1;
    bool f0 = (c == 1), f1 = (c == 2), f2 = (c == 0);
    unsigned long long lmask = (1ull << lane) - 1ull;

    unsigned long long b0 = __ballot(f0);
    unsigned long long b1 = __ballot(f1);
    unsigned long long b2 = __ballot(f2);
    int r0 = __popcll(b0 & lmask), r1 = __popcll(b1 & lmask), r2 = __popcll(b2 & lmask);
    if (lane == 0) {
      waveOff[0][wid] = __popcll(b0);
      waveOff[1][wid] = __popcll(b1);
      waveOff[2][wid] = __popcll(b2);
    }
    __syncthreads();
    if (tid == 0) {
      for (int t = 0; t < 3; ++t) {
        int s = runTot[t];
        for (int w = 0; w < nWaves; ++w) {
          int v = waveOff[t][w];
          waveOff[t][w] = s;
          s += v;
        }
        runTot[t] = s;
      }
    }
    __syncthreads();
    if (f0) { int p = waveOff[0][wid] + r0; if (p < FG_MAX) fgList[p] = i; }
    if (f1) { int p = waveOff[1][wid] + r1; if (p < BATCH)  bgList[p] = i; }
    if (f2) { int p = waveOff[2][wid] + r2; if (p < BATCH)  otList[p] = i; }
    __syncthreads();
  }

  int nFg = runTot[0], nBg = runTot[1];
  int fg_count = nFg < FG_MAX ? nFg : FG_MAX;
  int avail = BATCH - fg_count;
  int bg_count = nBg < avail ? nBg : avail;
  if (tid < BATCH) {
    int k;
    if (tid < fg_count) {
      k = fgList[tid];
    } else {
      int j = tid - fg_count;
      k = (j < nBg) ? bgList[j] : otList[j - nBg];  // argsort overflow path
    }
    keep[tid] = k;
  }
  if (tid == 0) { meta[0] = fg_count; meta[1] = bg_count; }
}

// ---------------------------------------------------------------------------
// Stage D: gather rois, bbox_transform, one-hot labels, scattered bbox targets.
// One block per output slot.
// ---------------------------------------------------------------------------
__global__ void output_kernel(const float* __restrict__ proposals, int Np,
                              const float* __restrict__ gt, int K,
                              const int* __restrict__ assign,
                              const int* __restrict__ gt_cls,
                              const int* __restrict__ keep,
                              const int* __restrict__ meta,
                              float* __restrict__ out, int C) {
#pragma clang fp contract(off)
  int s = blockIdx.x;
  int tid = threadIdx.x;
  __shared__ float sroi[4];
  __shared__ float st[4];
  __shared__ int   slbl;
  if (tid == 0) {
    int fg_count = meta[0];
    int k = keep[s];
    const float* r = (k < Np) ? (proposals + (size_t)k * 4)
                              : (gt + (size_t)(k - Np) * 4);
    float ex0 = r[0], ey0 = r[1], ex1 = r[2], ey1 = r[3];
    int a = assign[k];
    const float* g = gt + (size_t)a * 4;
    float ew = ex1 - ex0 + 1.0f, eh = ey1 - ey0 + 1.0f;
    float ecx = ex0 + 0.5f * ew, ecy = ey0 + 0.5f * eh;
    float gw = g[2] - g[0] + 1.0f, gh = g[3] - g[1] + 1.0f;
    float gcx = g[0] + 0.5f * gw, gcy = g[1] + 0.5f * gh;
    st[0] = (gcx - ecx) / ew;
    st[1] = (gcy - ecy) / eh;
    st[2] = logf(gw / ew);
    st[3] = logf(gh / eh);
    sroi[0] = ex0; sroi[1] = ey0; sroi[2] = ex1; sroi[3] = ey1;
    slbl = (s < fg_count) ? gt_cls[a] : 0;   // is_fg_slot implies valid
  }
  __syncthreads();
  int lbl = slbl;

  if (tid < 4) out[s * 4 + tid] = sroi[tid];

  float* labels = out + BATCH * 4 + (size_t)s * C;
  for (int c = tid; c < C; c += blockDim.x)
    labels[c] = (c == lbl) ? 1.0f : 0.0f;

  float* bb = out + BATCH * 4 + (size_t)BATCH * C + (size_t)s * 4 * C;
  for (int c4 = tid; c4 < 4 * C; c4 += blockDim.x) {
    int c = c4 >> 2, j = c4 & 3;
    bb[c4] = (lbl > 0 && c == lbl) ? st[j] : 0.0f;
  }
}

// ---------------------------------------------------------------------------
extern "C" void kernel_launch(void* const* d_in, const int* in_sizes, int n_in,
                              void* d_out, int out_size, void* d_ws, size_t ws_size,
                              hipStream_t stream) {
  const float* proposals = (const float*)d_in[0];
  const float* gt_labels = (const float*)d_in[1];
  const float* gt_boxes  = (const float*)d_in[2];
  int Np = in_sizes[0] / 4;
  int K  = in_sizes[2] / 4;
  int C  = in_sizes[1] / K;
  int n_total = Np + K;

  char* ws = (char*)d_ws;
  int* assign = (int*)ws;  ws += sizeof(int) * (size_t)n_total;
  int* code   = (int*)ws;  ws += sizeof(int) * (size_t)n_total;
  int* gt_cls = (int*)ws;  ws += sizeof(int) * (size_t)((K + 15) & ~15);
  int* keep   = (int*)ws;  ws += sizeof(int) * BATCH;
  int* meta   = (int*)ws;

  int nTiles = (K + 15) / 16;
  int gA = (nTiles + 31) / 32;            // 32 waves (1024 threads) per block
  gtcls_wmma_kernel<<<gA, 1024, 0, stream>>>(gt_labels, K, C, gt_cls);
  iou_kernel<<<(n_total + 255) / 256, 256, 0, stream>>>(proposals, Np, gt_boxes,
                                                        K, assign, code);
  select_kernel<<<1, 1024, 0, stream>>>(code, n_total, keep, meta);
  output_kernel<<<BATCH, 256, 0, stream>>>(proposals, Np, gt_boxes, K, assign,
                                           gt_cls, keep, meta, (float*)d_out, C);
}